// StatelessNTM_17798344475183
// MI455X (gfx1250) — compile-verified
//
#include <hip/hip_runtime.h>
#include <math.h>

#define EPSF 1e-8f
#define RB 512   // reduction grid blocks

typedef __attribute__((ext_vector_type(2))) float v2f;
typedef __attribute__((ext_vector_type(4))) float f4;
typedef __attribute__((ext_vector_type(8))) float v8f;

// workspace float offsets
#define OFF_H   32
#define OFF_KN  288
#define OFF_KW  544
#define OFF_E   800
#define OFF_A   1056
#define OFF_BIG 2048

__device__ __forceinline__ float softplusf(float x){
  return fmaxf(x, 0.f) + log1pf(expf(-fabsf(x)));
}
__device__ __forceinline__ float sigmoidf(float x){
  return 1.f / (1.f + expf(-x));
}

// ---------------------------------------------------------------------------
// Kernel 1: controller + head linears + derived head parameters (1 block)
// ws[0..5]  = beta_r, g_r, s_r0, s_r1, s_r2, gamma_r
// ws[6..11] = beta_w, g_w, s_w0, s_w1, s_w2, gamma_w
// ws[OFF_H..]=h(200), OFF_KN: kn(256), OFF_KW: kw(256), OFF_E: e(256), OFF_A: a(256)
// ---------------------------------------------------------------------------
__global__ __launch_bounds__(256) void k_head(
    const float* __restrict__ x, const float* __restrict__ read_prev,
    const float* __restrict__ reads_bias,
    const float* __restrict__ W1, const float* __restrict__ b1,
    const float* __restrict__ Wr, const float* __restrict__ br,
    const float* __restrict__ Ww, const float* __restrict__ bw,
    float* __restrict__ ws)
{
  __shared__ float s_xc[512];
  __shared__ float s_h[200];
  __shared__ float s_o[774];
  __shared__ float s_red[256];
  const int t = threadIdx.x;
  s_xc[t]       = x[t];
  s_xc[256 + t] = read_prev[t] + reads_bias[t];
  __syncthreads();
  if (t < 200){
    float acc = b1[t];
    for (int i = 0; i < 512; i++) acc = fmaf(s_xc[i], W1[i*200 + t], acc);
    float h = fmaxf(acc, 0.f);
    s_h[t] = h;
    ws[OFF_H + t] = h;
  }
  __syncthreads();
  // read head: orh = h @ Wr + br  (262)
  for (int j = t; j < 262; j += 256){
    float acc = br[j];
    for (int i = 0; i < 200; i++) acc = fmaf(s_h[i], Wr[i*262 + j], acc);
    s_o[j] = acc;
  }
  __syncthreads();
  {
    float kr = tanhf(s_o[t]);
    s_red[t] = kr * kr;
    __syncthreads();
    for (int s = 128; s > 0; s >>= 1){ if (t < s) s_red[t] += s_red[t + s]; __syncthreads(); }
    float nrm = sqrtf(s_red[0]);
    ws[OFF_KN + t] = kr / (nrm + EPSF);
    if (t == 0){
      ws[0] = softplusf(s_o[256]);
      ws[1] = sigmoidf(s_o[257]);
      float m  = fmaxf(s_o[258], fmaxf(s_o[259], s_o[260]));
      float e0 = expf(s_o[258]-m), e1 = expf(s_o[259]-m), e2 = expf(s_o[260]-m);
      float se = e0 + e1 + e2;
      ws[2] = e0/se; ws[3] = e1/se; ws[4] = e2/se;
      ws[5] = 1.f + softplusf(s_o[261]);
    }
  }
  __syncthreads();
  // write head: owh = h @ Ww + bw  (774)
  for (int j = t; j < 774; j += 256){
    float acc = bw[j];
    for (int i = 0; i < 200; i++) acc = fmaf(s_h[i], Ww[i*774 + j], acc);
    s_o[j] = acc;
  }
  __syncthreads();
  {
    float kw = tanhf(s_o[t]);
    s_red[t] = kw * kw;
    __syncthreads();
    for (int s = 128; s > 0; s >>= 1){ if (t < s) s_red[t] += s_red[t + s]; __syncthreads(); }
    float nrm = sqrtf(s_red[0]);
    ws[OFF_KW + t] = kw / (nrm + EPSF);
    ws[OFF_E + t]  = sigmoidf(s_o[262 + t]);
    ws[OFF_A + t]  = tanhf(s_o[518 + t]);
    if (t == 0){
      ws[6] = softplusf(s_o[256]);
      ws[7] = sigmoidf(s_o[257]);
      float m  = fmaxf(s_o[258], fmaxf(s_o[259], s_o[260]));
      float e0 = expf(s_o[258]-m), e1 = expf(s_o[259]-m), e2 = expf(s_o[260]-m);
      float se = e0 + e1 + e2;
      ws[8] = e0/se; ws[9] = e1/se; ws[10] = e2/se;
      ws[11] = 1.f + softplusf(s_o[261]);
    }
  }
}

// ---------------------------------------------------------------------------
// Kernel 2: scan over M. One wave handles a 16-row tile.
// D = A(16x4 rows of M) x B(4x16; col0=kn chunk, col1=kw chunk) accumulated
// over 64 K-chunks via V_WMMA_F32_16X16X4_F32; row norms via VALU.
// score = beta * dot / (||row|| + eps)
// ---------------------------------------------------------------------------
#if defined(__gfx1250__) && __has_builtin(__builtin_amdgcn_wmma_f32_16x16x4_f32)
#define USE_WMMA_F32 1
#endif

__global__ __launch_bounds__(256) void k_score(
    const float* __restrict__ M, const float* __restrict__ ws,
    float* __restrict__ score_r, float* __restrict__ score_w)
{
  const float* kn = ws + OFF_KN;
  const float* kw = ws + OFF_KW;
  const float beta_r = ws[0];
  const float beta_w = ws[6];
  const int lane = threadIdx.x & 31;
  const int wid  = threadIdx.x >> 5;
  const int rowbase = (blockIdx.x * 8 + wid) * 16;
  const int r  = lane & 15;          // A row index / B column index
  const int hi = (lane >> 4) & 1;    // K offset {0,1} vs {2,3}
  const float* rowp = M + (size_t)(rowbase + r) * 256 + 2*hi;

#ifdef USE_WMMA_F32
  const float* keyp = (r == 0) ? kn : kw;   // only used for r<2
  v8f acc = {0.f,0.f,0.f,0.f,0.f,0.f,0.f,0.f};
  float nrm = 0.f;
  for (int c = 0; c < 64; c++){
    float ax = rowp[4*c];
    float ay = rowp[4*c + 1];
    v2f A; A[0] = ax; A[1] = ay;
    float b0 = 0.f, b1 = 0.f;
    if (r < 2){ b0 = keyp[4*c + 2*hi]; b1 = keyp[4*c + 2*hi + 1]; }
    v2f B; B[0] = b0; B[1] = b1;
    acc = __builtin_amdgcn_wmma_f32_16x16x4_f32(
        false, A, false, B, (short)0, acc, false, false);
    nrm = fmaf(ax, ax, fmaf(ay, ay, nrm));
  }
  float fulln = nrm + __shfl_xor(nrm, 16, 32);
  float nv[8];
  #pragma unroll
  for (int j = 0; j < 8; j++) nv[j] = __shfl(fulln, hi*8 + j, 32);
  if (r < 2){
    float beta = (r == 0) ? beta_r : beta_w;
    float* op = ((r == 0) ? score_r : score_w) + rowbase + hi*8;
    #pragma unroll
    for (int j = 0; j < 8; j++) op[j] = beta * acc[j] / (sqrtf(nv[j]) + EPSF);
  }
#else
  // shuffle-reduce fallback (also what the host-side parse sees)
  float pdr = 0.f, pdw = 0.f, nrm = 0.f;
  for (int c = 0; c < 64; c++){
    float ax = rowp[4*c];
    float ay = rowp[4*c + 1];
    int k0 = 4*c + 2*hi;
    pdr = fmaf(ax, kn[k0], fmaf(ay, kn[k0+1], pdr));
    pdw = fmaf(ax, kw[k0], fmaf(ay, kw[k0+1], pdw));
    nrm = fmaf(ax, ax, fmaf(ay, ay, nrm));
  }
  float dr = pdr + __shfl_xor(pdr, 16, 32);
  float dw = pdw + __shfl_xor(pdw, 16, 32);
  float fn = nrm + __shfl_xor(nrm, 16, 32);
  if (lane < 16){
    float inv = 1.f / (sqrtf(fn) + EPSF);
    score_r[rowbase + r] = beta_r * dr * inv;
    score_w[rowbase + r] = beta_w * dw * inv;
  }
#endif
}

// ---------------------------------------------------------------------------
// Kernel 3: per-block max partials of both score arrays
// ---------------------------------------------------------------------------
__global__ __launch_bounds__(256) void k_max(
    const float* __restrict__ sr, const float* __restrict__ sw,
    float* __restrict__ pmr, float* __restrict__ pmw, int n)
{
  __shared__ float l1[256], l2[256];
  const int t = threadIdx.x;
  float mr = -3.402823466e38f, mw = -3.402823466e38f;
  for (int i = blockIdx.x*256 + t; i < n; i += gridDim.x*256){
    mr = fmaxf(mr, sr[i]); mw = fmaxf(mw, sw[i]);
  }
  l1[t] = mr; l2[t] = mw; __syncthreads();
  for (int s = 128; s > 0; s >>= 1){
    if (t < s){ l1[t] = fmaxf(l1[t], l1[t+s]); l2[t] = fmaxf(l2[t], l2[t+s]); }
    __syncthreads();
  }
  if (t == 0){ pmr[blockIdx.x] = l1[0]; pmw[blockIdx.x] = l2[0]; }
}

// ---------------------------------------------------------------------------
// Kernel 4: finish max (redundantly per block), exp, per-block sum partials
// ---------------------------------------------------------------------------
__global__ __launch_bounds__(256) void k_exp(
    const float* __restrict__ sr, const float* __restrict__ sw,
    const float* __restrict__ pmr, const float* __restrict__ pmw,
    float* __restrict__ er, float* __restrict__ ew,
    float* __restrict__ psr, float* __restrict__ psw, int n)
{
  __shared__ float l1[256], l2[256];
  const int t = threadIdx.x;
  float mr = -3.402823466e38f, mw = -3.402823466e38f;
  for (int i = t; i < RB; i += 256){ mr = fmaxf(mr, pmr[i]); mw = fmaxf(mw, pmw[i]); }
  l1[t] = mr; l2[t] = mw; __syncthreads();
  for (int s = 128; s > 0; s >>= 1){
    if (t < s){ l1[t] = fmaxf(l1[t], l1[t+s]); l2[t] = fmaxf(l2[t], l2[t+s]); }
    __syncthreads();
  }
  mr = l1[0]; mw = l2[0];
  __syncthreads();
  float ar = 0.f, aw = 0.f;
  for (int i = blockIdx.x*256 + t; i < n; i += gridDim.x*256){
    float a = expf(sr[i] - mr); er[i] = a; ar += a;
    float b = expf(sw[i] - mw); ew[i] = b; aw += b;
  }
  l1[t] = ar; l2[t] = aw; __syncthreads();
  for (int s = 128; s > 0; s >>= 1){
    if (t < s){ l1[t] += l1[t+s]; l2[t] += l2[t+s]; }
    __syncthreads();
  }
  if (t == 0){ psr[blockIdx.x] = l1[0]; psw[blockIdx.x] = l2[0]; }
}

// ---------------------------------------------------------------------------
// Kernel 5: finish sums; fused interpolation + circular shift + sharpen;
// per-block sum partials of wp
// ---------------------------------------------------------------------------
__global__ __launch_bounds__(256) void k_shift(
    const float* __restrict__ er, const float* __restrict__ ew,
    const float* __restrict__ prevr, const float* __restrict__ prevw,
    const float* __restrict__ psr, const float* __restrict__ psw,
    const float* __restrict__ ws,
    float* __restrict__ wpr, float* __restrict__ wpw,
    float* __restrict__ ppr, float* __restrict__ ppw, int n)
{
  __shared__ float l1[256], l2[256];
  const int t = threadIdx.x;
  float ar = 0.f, aw = 0.f;
  for (int i = t; i < RB; i += 256){ ar += psr[i]; aw += psw[i]; }
  l1[t] = ar; l2[t] = aw; __syncthreads();
  for (int s = 128; s > 0; s >>= 1){
    if (t < s){ l1[t] += l1[t+s]; l2[t] += l2[t+s]; }
    __syncthreads();
  }
  const float invSR = 1.f / l1[0];
  const float invSW = 1.f / l2[0];
  __syncthreads();
  const float gr = ws[1],  s0r = ws[2], s1r = ws[3], s2r = ws[4], gar = ws[5];
  const float gw = ws[7],  s0w = ws[8], s1w = ws[9], s2w = ws[10], gaw = ws[11];
  float accr = 0.f, accw = 0.f;
  for (int i = blockIdx.x*256 + t; i < n; i += gridDim.x*256){
    int ip = (i + 1 == n) ? 0 : i + 1;
    int im = (i == 0) ? n - 1 : i - 1;
    // read head
    {
      float wg_ip = gr * er[ip] * invSR + (1.f - gr) * prevr[ip];
      float wg_i  = gr * er[i ] * invSR + (1.f - gr) * prevr[i ];
      float wg_im = gr * er[im] * invSR + (1.f - gr) * prevr[im];
      float wt = s0r*wg_ip + s1r*wg_i + s2r*wg_im;
      float p  = powf(wt, gar);
      wpr[i] = p; accr += p;
    }
    // write head
    {
      float wg_ip = gw * ew[ip] * invSW + (1.f - gw) * prevw[ip];
      float wg_i  = gw * ew[i ] * invSW + (1.f - gw) * prevw[i ];
      float wg_im = gw * ew[im] * invSW + (1.f - gw) * prevw[im];
      float wt = s0w*wg_ip + s1w*wg_i + s2w*wg_im;
      float p  = powf(wt, gaw);
      wpw[i] = p; accw += p;
    }
  }
  l1[t] = accr; l2[t] = accw; __syncthreads();
  for (int s = 128; s > 0; s >>= 1){
    if (t < s){ l1[t] += l1[t+s]; l2[t] += l2[t+s]; }
    __syncthreads();
  }
  if (t == 0){ ppr[blockIdx.x] = l1[0]; ppw[blockIdx.x] = l2[0]; }
}

// ---------------------------------------------------------------------------
// Kernel 6: finish wp sums; normalize -> final rw, ww written to d_out
// ---------------------------------------------------------------------------
__global__ __launch_bounds__(256) void k_norm(
    const float* __restrict__ wpr, const float* __restrict__ wpw,
    const float* __restrict__ ppr, const float* __restrict__ ppw,
    float* __restrict__ rw_out, float* __restrict__ ww_out, int n)
{
  __shared__ float l1[256], l2[256];
  const int t = threadIdx.x;
  float ar = 0.f, aw = 0.f;
  for (int i = t; i < RB; i += 256){ ar += ppr[i]; aw += ppw[i]; }
  l1[t] = ar; l2[t] = aw; __syncthreads();
  for (int s = 128; s > 0; s >>= 1){
    if (t < s){ l1[t] += l1[t+s]; l2[t] += l2[t+s]; }
    __syncthreads();
  }
  const float invR = 1.f / (l1[0] + EPSF);
  const float invW = 1.f / (l2[0] + EPSF);
  for (int i = blockIdx.x*256 + t; i < n; i += gridDim.x*256){
    rw_out[i] = wpr[i] * invR;
    ww_out[i] = wpw[i] * invW;
  }
}

// ---------------------------------------------------------------------------
// Kernel 7: fused memory update M_new = M*(1 - ww^T e) + ww^T a  (NT stores)
// and read-vector partials: pread[block][d] = sum_i rw[i]*M[i][d]
// block handles 128 rows; thread t -> cols [4*(t%64) .. +3], row group t/64
// ---------------------------------------------------------------------------
__global__ __launch_bounds__(256) void k_update(
    const float* __restrict__ M,
    const float* __restrict__ rw, const float* __restrict__ ww,
    const float* __restrict__ ws,
    float* __restrict__ Mnew, float* __restrict__ pread)
{
  __shared__ float ls[4*256];
  const int t   = threadIdx.x;
  const int sub = t & 63;
  const int rs  = t >> 6;
  const int c0  = sub * 4;
  const f4 e4 = *(const f4*)(ws + OFF_E + c0);
  const f4 a4 = *(const f4*)(ws + OFF_A + c0);
  float ar0 = 0.f, ar1 = 0.f, ar2 = 0.f, ar3 = 0.f;
  const int base = blockIdx.x * 128;
  for (int rr = rs; rr < 128; rr += 4){
    const int row = base + rr;
    const float wwi = ww[row];
    const float rwi = rw[row];
    const f4 m = *(const f4*)(M + (size_t)row * 256 + c0);
    f4 o;
    o[0] = m[0] * (1.f - wwi*e4[0]) + wwi*a4[0];
    o[1] = m[1] * (1.f - wwi*e4[1]) + wwi*a4[1];
    o[2] = m[2] * (1.f - wwi*e4[2]) + wwi*a4[2];
    o[3] = m[3] * (1.f - wwi*e4[3]) + wwi*a4[3];
    __builtin_nontemporal_store(o, (f4*)(Mnew + (size_t)row * 256 + c0));
    ar0 = fmaf(rwi, m[0], ar0);
    ar1 = fmaf(rwi, m[1], ar1);
    ar2 = fmaf(rwi, m[2], ar2);
    ar3 = fmaf(rwi, m[3], ar3);
  }
  ls[rs*256 + c0 + 0] = ar0;
  ls[rs*256 + c0 + 1] = ar1;
  ls[rs*256 + c0 + 2] = ar2;
  ls[rs*256 + c0 + 3] = ar3;
  __syncthreads();
  if (rs == 0){
    #pragma unroll
    for (int k = 0; k < 4; k++){
      int col = c0 + k;
      pread[(size_t)blockIdx.x * 256 + col] =
          ls[col] + ls[256 + col] + ls[512 + col] + ls[768 + col];
    }
  }
}

// ---------------------------------------------------------------------------
// Kernel 8: finish read vector; output MLP; write out + read (1 block)
// ---------------------------------------------------------------------------
__global__ __launch_bounds__(256) void k_out(
    const float* __restrict__ ws, const float* __restrict__ pread,
    const float* __restrict__ Wo1, const float* __restrict__ bo1,
    const float* __restrict__ Wo2, const float* __restrict__ bo2,
    float* __restrict__ out_v, float* __restrict__ read_v, int nblk)
{
  __shared__ float cat[456];
  __shared__ float h2[100];
  const int t = threadIdx.x;
  if (t < 200) cat[t] = ws[OFF_H + t];
  float s = 0.f;
  for (int b = 0; b < nblk; b++) s += pread[(size_t)b * 256 + t];
  cat[200 + t] = s;
  read_v[t] = s;
  __syncthreads();
  if (t < 100){
    float acc = bo1[t];
    for (int i = 0; i < 456; i++) acc = fmaf(cat[i], Wo1[i*100 + t], acc);
    h2[t] = fmaxf(acc, 0.f);
  }
  __syncthreads();
  if (t < 64){
    float acc = bo2[t];
    for (int j = 0; j < 100; j++) acc = fmaf(h2[j], Wo2[j*64 + t], acc);
    out_v[t] = acc;
  }
}

// ---------------------------------------------------------------------------
extern "C" void kernel_launch(void* const* d_in, const int* in_sizes, int n_in,
                              void* d_out, int out_size, void* d_ws, size_t ws_size,
                              hipStream_t stream)
{
  (void)n_in; (void)out_size; (void)ws_size;
  const float* x          = (const float*)d_in[0];
  const float* M          = (const float*)d_in[1];
  const float* read_w     = (const float*)d_in[2];
  const float* write_w    = (const float*)d_in[3];
  const float* read_prev  = (const float*)d_in[4];
  const float* reads_bias = (const float*)d_in[5];
  const float* W1  = (const float*)d_in[6];
  const float* b1  = (const float*)d_in[7];
  const float* Wr  = (const float*)d_in[8];
  const float* br  = (const float*)d_in[9];
  const float* Ww  = (const float*)d_in[10];
  const float* bw  = (const float*)d_in[11];
  const float* Wo1 = (const float*)d_in[12];
  const float* bo1 = (const float*)d_in[13];
  const float* Wo2 = (const float*)d_in[14];
  const float* bo2 = (const float*)d_in[15];

  const int N = in_sizes[2];            // 131072 memory rows

  float* o      = (float*)d_out;        // out(64) | M_new(N*256) | rw(N) | ww(N) | read(256)
  float* out_v  = o;
  float* Mnew   = o + 64;
  float* rw_v   = Mnew + (size_t)N * 256;
  float* ww_v   = rw_v + N;
  float* read_v = ww_v + N;

  float* ws     = (float*)d_ws;
  float* sc_r   = ws + OFF_BIG;
  float* sc_w   = sc_r + N;
  float* es_r   = sc_w + N;
  float* es_w   = es_r + N;
  float* wp_r   = es_w + N;
  float* wp_w   = wp_r + N;
  float* pmax_r = wp_w + N;
  float* pmax_w = pmax_r + RB;
  float* psum_r = pmax_w + RB;
  float* psum_w = psum_r + RB;
  float* ppow_r = psum_w + RB;
  float* ppow_w = ppow_r + RB;
  float* pread  = ppow_w + RB;          // (N/128) * 256 floats

  const int NB = N / 128;               // 1024 blocks over M rows

  k_head  <<<1,   256, 0, stream>>>(x, read_prev, reads_bias, W1, b1, Wr, br, Ww, bw, ws);
  k_score <<<NB,  256, 0, stream>>>(M, ws, sc_r, sc_w);
  k_max   <<<RB,  256, 0, stream>>>(sc_r, sc_w, pmax_r, pmax_w, N);
  k_exp   <<<RB,  256, 0, stream>>>(sc_r, sc_w, pmax_r, pmax_w, es_r, es_w, psum_r, psum_w, N);
  k_shift <<<RB,  256, 0, stream>>>(es_r, es_w, read_w, write_w, psum_r, psum_w, ws,
                                    wp_r, wp_w, ppow_r, ppow_w, N);
  k_norm  <<<RB,  256, 0, stream>>>(wp_r, wp_w, ppow_r, ppow_w, rw_v, ww_v, N);
  k_update<<<NB,  256, 0, stream>>>(M, rw_v, ww_v, ws, Mnew, pread);
  k_out   <<<1,   256, 0, stream>>>(ws, pread, Wo1, bo1, Wo2, bo2, out_v, read_v, NB);
}